// SplineCNN_86483461472644
// MI455X (gfx1250) — compile-verified
//
#include <hip/hip_runtime.h>
#include <hip/hip_bf16.h>
#include <math.h>

typedef float v2f __attribute__((ext_vector_type(2)));
typedef float v8f __attribute__((ext_vector_type(8)));

#define D 128          // feature dim (in == out == 128)
#define KS 5
#define KTOT 25        // KS*KS
#define OUTW 384       // D * 3 concat output width

// ---------------------------------------------------------------------------
// zero fill (graph-capture safe)
// ---------------------------------------------------------------------------
__global__ void zero_kernel(float* __restrict__ p, long long n) {
    long long i = (long long)blockIdx.x * blockDim.x + threadIdx.x;
    if (i < n) p[i] = 0.0f;
}

// ---------------------------------------------------------------------------
// degree: deg[dst] += 1 per edge (dst identical for both layers -> run once)
// ---------------------------------------------------------------------------
__global__ void deg_kernel(const int* __restrict__ dst, float* __restrict__ deg, int n_edges) {
    int e = blockIdx.x * blockDim.x + threadIdx.x;
    if (e < n_edges) atomicAdd(&deg[dst[e]], 1.0f);
}

// ---------------------------------------------------------------------------
// l2norm of raw x into out columns [0,128)   (wave per node, 4 chans per lane)
// ---------------------------------------------------------------------------
__global__ void l2norm_x_kernel(const float* __restrict__ x, float* __restrict__ out, int n_nodes) {
    int wave = blockIdx.x * (blockDim.x >> 5) + (threadIdx.x >> 5);
    if (wave >= n_nodes) return;
    int lane = threadIdx.x & 31;
    const float* row = x + (size_t)wave * D;
    float v0 = row[lane], v1 = row[lane + 32], v2 = row[lane + 64], v3 = row[lane + 96];
    float ss = v0 * v0 + v1 * v1 + v2 * v2 + v3 * v3;
    #pragma unroll
    for (int off = 16; off > 0; off >>= 1) ss += __shfl_xor(ss, off, 32);
    float inv = 1.0f / fmaxf(sqrtf(ss), 1e-12f);
    float* o = out + (size_t)wave * OUTW;
    o[lane] = v0 * inv; o[lane + 32] = v1 * inv;
    o[lane + 64] = v2 * inv; o[lane + 96] = v3 * inv;
}

// ---------------------------------------------------------------------------
// Xk chunk GEMM via fp32 WMMA:  xk[n, j*128+o] = sum_i xin[n,i] * w[k0+j,i,o]
// Register-blocked: one wave computes a 16x64 output slab (4 N-tiles sharing
// the A operand), K=128 swept as 32 steps of 4 chained v_wmma_f32_16x16x4_f32.
// A layout (f32 16x4): lane m=L&15; VGPR0=K{0|2}, VGPR1=K{1|3} by lane half.
// B layout (f32 4x16): col=L&15;   VGPR0=K{0|2}, VGPR1=K{1|3} by lane half.
// C/D: 8 VGPRs, row = v + 8*(L>=16), col = L&15.
// ---------------------------------------------------------------------------
__global__ void xk_gemm_kernel(const float* __restrict__ xin, const float* __restrict__ w,
                               float* __restrict__ xk, int k0, int kc, int n_nodes) {
    const int Wd = kc * D;                    // chunk output width
    const int tiles_n = kc * 2;               // 64-wide slabs (2 per k-submatrix)
    const int tiles_m = n_nodes >> 4;
    const int wave = blockIdx.x * (blockDim.x >> 5) + (threadIdx.x >> 5);
    if (wave >= tiles_m * tiles_n) return;
    const int tm  = wave / tiles_n;
    const int tnb = wave - tm * tiles_n;
    const int lane = threadIdx.x & 31;
    const int m  = lane & 15;                 // A row within tile
    const int h2 = (lane >> 4) << 1;          // 0 or 2: K sub-offset for lane half
    const int col  = lane & 15;               // B/D column within tile
    const int gcol = tnb * 64 + col;          // slab never straddles a k boundary
    const int j = gcol >> 7;                  // which k inside chunk
    const int o = gcol & (D - 1);             // output channel base
    const float* arow = xin + (size_t)(tm * 16 + m) * D;
    const float* bcol = w + ((size_t)(k0 + j) * D) * D + o;   // + i*D walks rows i
    v8f c0 = {}, c1 = {}, c2 = {}, c3 = {};
    #pragma unroll
    for (int kk = 0; kk < D; kk += 4) {
        v2f a = *(const v2f*)(arow + kk + h2);          // A[m][kk+h2 .. +1]
        const float* br0 = bcol + (size_t)(kk + h2) * D;      // B row kk+h2
        const float* br1 = br0 + D;                           // B row kk+h2+1
        v2f b0 = { br0[0],  br1[0]  };
        v2f b1 = { br0[16], br1[16] };
        v2f b2 = { br0[32], br1[32] };
        v2f b3 = { br0[48], br1[48] };
        c0 = __builtin_amdgcn_wmma_f32_16x16x4_f32(false, a, false, b0, (short)0, c0, false, false);
        c1 = __builtin_amdgcn_wmma_f32_16x16x4_f32(false, a, false, b1, (short)0, c1, false, false);
        c2 = __builtin_amdgcn_wmma_f32_16x16x4_f32(false, a, false, b2, (short)0, c2, false, false);
        c3 = __builtin_amdgcn_wmma_f32_16x16x4_f32(false, a, false, b3, (short)0, c3, false, false);
    }
    const int hh = (lane >> 4) << 3;          // 0 or 8: D-row offset for lane half
    float* drow = xk + (size_t)(tm * 16) * Wd + tnb * 64 + col;
    #pragma unroll
    for (int v = 0; v < 8; ++v) {
        float* r = drow + (size_t)(v + hh) * Wd;
        r[0]  = c0[v];
        r[16] = c1[v];
        r[32] = c2[v];
        r[48] = c3[v];
    }
}

// ---------------------------------------------------------------------------
// Edge gather + weighted scatter-add for spline combos in chunk [k0, k0+kc).
// One wave per edge; lane handles 4 channels; combos accumulated in registers
// then a single atomic add per channel.
// ---------------------------------------------------------------------------
__global__ void edge_kernel(const int* __restrict__ src, const int* __restrict__ dst,
                            const float* __restrict__ pseudo, const float* __restrict__ xk,
                            float* __restrict__ agg, int k0, int kc, int n_edges) {
    int wave = blockIdx.x * (blockDim.x >> 5) + (threadIdx.x >> 5);
    if (wave >= n_edges) return;
    int lane = threadIdx.x & 31;
    int s = src[wave], d = dst[wave];
    float v0 = pseudo[2 * wave] * (float)(KS - 1);
    float v1 = pseudo[2 * wave + 1] * (float)(KS - 1);
    float fl0 = fminf(fmaxf(floorf(v0), 0.0f), (float)(KS - 2));
    float fl1 = fminf(fmaxf(floorf(v1), 0.0f), (float)(KS - 2));
    int i0 = (int)fl0, i1 = (int)fl1;
    float f0 = v0 - fl0, f1 = v1 - fl1;
    const float* base = xk + (size_t)s * (kc * D);
    float a0 = 0.f, a1 = 0.f, a2 = 0.f, a3 = 0.f;
    bool any = false;
    #pragma unroll
    for (int by = 0; by < 2; ++by) {
        #pragma unroll
        for (int bx = 0; bx < 2; ++bx) {
            int wi = (i0 + bx) + KS * (i1 + by);
            if (wi >= k0 && wi < k0 + kc) {
                float wgt = (bx ? f0 : 1.0f - f0) * (by ? f1 : 1.0f - f1);
                const float* row = base + (size_t)(wi - k0) * D;
                a0 += wgt * row[lane];       a1 += wgt * row[lane + 32];
                a2 += wgt * row[lane + 64]; a3 += wgt * row[lane + 96];
                any = true;
            }
        }
    }
    if (any) {
        float* og = agg + (size_t)d * D;
        atomicAdd(og + lane, a0);       atomicAdd(og + lane + 32, a1);
        atomicAdd(og + lane + 64, a2); atomicAdd(og + lane + 96, a3);
    }
}

// ---------------------------------------------------------------------------
// finalize: h = agg/max(deg,1)+bias;  out[:,colbase:] = l2norm(h);  hout = relu(h)
// ---------------------------------------------------------------------------
__global__ void finalize_kernel(const float* __restrict__ agg, const float* __restrict__ deg,
                                const float* __restrict__ bias, float* __restrict__ out,
                                float* __restrict__ hout, int n_nodes, int colbase) {
    int wave = blockIdx.x * (blockDim.x >> 5) + (threadIdx.x >> 5);
    if (wave >= n_nodes) return;
    int lane = threadIdx.x & 31;
    const float* arow = agg + (size_t)wave * D;
    float invd = 1.0f / fmaxf(deg[wave], 1.0f);
    float v0 = arow[lane] * invd + bias[lane];
    float v1 = arow[lane + 32] * invd + bias[lane + 32];
    float v2 = arow[lane + 64] * invd + bias[lane + 64];
    float v3 = arow[lane + 96] * invd + bias[lane + 96];
    float ss = v0 * v0 + v1 * v1 + v2 * v2 + v3 * v3;
    #pragma unroll
    for (int off = 16; off > 0; off >>= 1) ss += __shfl_xor(ss, off, 32);
    float inv = 1.0f / fmaxf(sqrtf(ss), 1e-12f);
    float* o = out + (size_t)wave * OUTW + colbase;
    o[lane] = v0 * inv; o[lane + 32] = v1 * inv;
    o[lane + 64] = v2 * inv; o[lane + 96] = v3 * inv;
    float* hr = hout + (size_t)wave * D;
    hr[lane] = fmaxf(v0, 0.f);       hr[lane + 32] = fmaxf(v1, 0.f);
    hr[lane + 64] = fmaxf(v2, 0.f); hr[lane + 96] = fmaxf(v3, 0.f);
}

// ---------------------------------------------------------------------------
extern "C" void kernel_launch(void* const* d_in, const int* in_sizes, int n_in,
                              void* d_out, int out_size, void* d_ws, size_t ws_size,
                              hipStream_t stream) {
    (void)n_in; (void)out_size;
    const float* x    = (const float*)d_in[0];
    const int*   ei   = (const int*)d_in[1];
    const float* attr = (const float*)d_in[2];
    const float* w0   = (const float*)d_in[3];
    const float* b0   = (const float*)d_in[4];
    const float* w1   = (const float*)d_in[5];
    const float* b1   = (const float*)d_in[6];
    float* out = (float*)d_out;

    const int N = in_sizes[0] / D;
    const int E = in_sizes[1] / 2;
    const int* src = ei;
    const int* dst = ei + E;

    // workspace layout: agg[N*D] | deg[N] | h[N*D] | xk[N*D*KC]
    float* agg = (float*)d_ws;
    float* deg = agg + (size_t)N * D;
    float* h   = deg + N;
    float* xk  = h + (size_t)N * D;
    size_t fixed_f = (size_t)N * (2 * D + 1);
    size_t avail_f = (ws_size / 4 > fixed_f) ? (ws_size / 4 - fixed_f) : 0;
    int KC = (int)(avail_f / ((size_t)N * D));
    if (KC > KTOT) KC = KTOT;
    if (KC < 1) KC = 1;                      // minimum viable chunk

    // degree (once; dst identical for both layers)
    {
        long long n = N;
        zero_kernel<<<(unsigned)((n + 255) / 256), 256, 0, stream>>>(deg, n);
        deg_kernel<<<(E + 255) / 256, 256, 0, stream>>>(dst, deg, E);
    }
    // out[:, 0:128) = l2norm(x)
    l2norm_x_kernel<<<(N * 32 + 255) / 256, 256, 0, stream>>>(x, out, N);

    for (int L = 0; L < 2; ++L) {
        const float* xin  = (L == 0) ? x : h;
        const float* w    = (L == 0) ? w0 : w1;
        const float* bias = (L == 0) ? b0 : b1;
        {
            long long n = (long long)N * D;
            zero_kernel<<<(unsigned)((n + 255) / 256), 256, 0, stream>>>(agg, n);
        }
        for (int k0 = 0; k0 < KTOT; k0 += KC) {
            int kc = (KTOT - k0 < KC) ? (KTOT - k0) : KC;
            long long tiles = (long long)(N >> 4) * (kc * 2);        // 16x64 slabs
            long long gemm_blocks = (tiles + 3) / 4;                 // 4 waves / 128-thr block
            xk_gemm_kernel<<<(unsigned)gemm_blocks, 128, 0, stream>>>(xin, w, xk, k0, kc, N);
            long long eb = ((long long)E + 7) / 8;                   // 8 waves / 256-thr block
            edge_kernel<<<(unsigned)eb, 256, 0, stream>>>(src, dst, attr, xk, agg, k0, kc, E);
        }
        finalize_kernel<<<(N * 32 + 255) / 256, 256, 0, stream>>>(agg, deg, bias, out, h,
                                                                  N, D * (L + 1));
    }
}